// SimpleGNN_19645180411919
// MI455X (gfx1250) — compile-verified
//
#include <hip/hip_runtime.h>

typedef __attribute__((ext_vector_type(2))) float v2f;
typedef __attribute__((ext_vector_type(8))) float v8f;

#define B_ 4
#define N_ 512
#define D_ 128
#define L_ 2
#define M_ (B_ * N_)   // 2048 rows in the flattened [B*N, D] view
#define TI 8           // i-rows per pairwise block
#define JT 64          // j-chunk per pairwise block

// ---------------------------------------------------------------------------
// GEMM via V_WMMA_F32_16X16X4_F32:
//   out[m][o] = bias[o] + sum_d X[m][d] * W[o*wstride + woff + d]
// One wave per 16x16 C tile; K stepped by 4 (32 WMMAs per tile).
// ---------------------------------------------------------------------------
__global__ __launch_bounds__(32)
void wmma_gemm_bias(const float* __restrict__ X, const float* __restrict__ W,
                    int wstride, int woff, const float* __restrict__ bias,
                    float* __restrict__ out) {
  const int lane = threadIdx.x;
  const int l16  = lane & 15;
  const int half = lane >> 4;          // 0: K lanes {k,k+1}; 1: {k+2,k+3}
  const int tm = blockIdx.y;
  const int tn = blockIdx.x;
  const float* xrow = X + (tm * 16 + l16) * D_;
  const float* wrow = W + (tn * 16 + l16) * wstride + woff;
  v8f c = {};
#pragma unroll
  for (int kk = 0; kk < D_ / 4; ++kk) {
    const int kb = kk * 4 + half * 2;
    v2f a = *(const v2f*)(xrow + kb);   // A 16x4 fragment (ISA 32-bit A layout)
    v2f b = *(const v2f*)(wrow + kb);   // B 4x16 fragment (W row-major == B^T)
    c = __builtin_amdgcn_wmma_f32_16x16x4_f32(
        false, a, false, b, (short)0, c, false, false);
  }
  const float bv = bias ? bias[tn * 16 + l16] : 0.f;
#pragma unroll
  for (int r = 0; r < 8; ++r) {
    const int m = tm * 16 + half * 8 + r;        // C layout: VGPR r -> M = 8*half + r
    out[m * D_ + tn * 16 + l16] = c[r] + bv;
  }
}

// ---------------------------------------------------------------------------
// Final fused GEMM: h[m][o] += S[m][:]·w2[o][:] + rowsum[m]*b2[o]   (in place)
// ---------------------------------------------------------------------------
__global__ __launch_bounds__(32)
void wmma_gemm_final(const float* __restrict__ S, const float* __restrict__ W2,
                     const float* __restrict__ b2, const float* __restrict__ rowsum,
                     float* __restrict__ h) {
  const int lane = threadIdx.x;
  const int l16  = lane & 15;
  const int half = lane >> 4;
  const int tm = blockIdx.y;
  const int tn = blockIdx.x;
  const float* xrow = S + (tm * 16 + l16) * D_;
  const float* wrow = W2 + (tn * 16 + l16) * D_;
  v8f c = {};
#pragma unroll
  for (int kk = 0; kk < D_ / 4; ++kk) {
    const int kb = kk * 4 + half * 2;
    v2f a = *(const v2f*)(xrow + kb);
    v2f b = *(const v2f*)(wrow + kb);
    c = __builtin_amdgcn_wmma_f32_16x16x4_f32(
        false, a, false, b, (short)0, c, false, false);
  }
  const float bv = b2[tn * 16 + l16];
#pragma unroll
  for (int r = 0; r < 8; ++r) {
    const int m = tm * 16 + half * 8 + r;
    const int idx = m * D_ + tn * 16 + l16;
    h[idx] = h[idx] + c[r] + rowsum[m] * bv;
  }
}

// ---------------------------------------------------------------------------
// Pairwise weighted reduction:
//   S[b,i,o] = sum_j adj[b,i,j] * relu(ai[b,i,o] + aj[b,j,o])
// Block: 256 threads = (o in [0,128)) x (ip in {0,1}); TI=8 i-rows, JT=64 j-chunk.
// aj tile staged transposed [o][jj] with pad 65 -> conflict-free ds loads;
// adj tile reads are wave-uniform broadcasts.
// ---------------------------------------------------------------------------
__global__ __launch_bounds__(256)
void pairwise_reduce(const float* __restrict__ ai, const float* __restrict__ aj,
                     const float* __restrict__ adj, float* __restrict__ S) {
  __shared__ float ajT[D_ * (JT + 1)];   // ~33 KB
  __shared__ float adjT[TI * JT];        // 2 KB
  const int b  = blockIdx.y;
  const int i0 = blockIdx.x * TI;
  const int tid = threadIdx.x;
  const int o  = tid & (D_ - 1);
  const int ip = tid >> 7;               // 0 or 1 (wave-uniform)

  float acc[TI / 2];
  float aiv[TI / 2];
#pragma unroll
  for (int r = 0; r < TI / 2; ++r) {
    acc[r] = 0.f;
    aiv[r] = ai[((b * N_) + i0 + ip * (TI / 2) + r) * D_ + o];
  }

  for (int j0 = 0; j0 < N_; j0 += JT) {
    __syncthreads();
    for (int t = tid; t < JT * D_; t += 256) {      // coalesced global read
      const int jj = t >> 7, oo = t & (D_ - 1);
      ajT[oo * (JT + 1) + jj] = aj[((b * N_) + j0 + jj) * D_ + oo];
    }
    for (int t = tid; t < TI * JT; t += 256) {
      const int ii = t >> 6, jj = t & (JT - 1);
      adjT[ii * JT + jj] = adj[((b * N_) + i0 + ii) * N_ + j0 + jj];
    }
    __syncthreads();
#pragma unroll 4
    for (int jj = 0; jj < JT; ++jj) {
      const float ajv = ajT[o * (JT + 1) + jj];     // bank-conflict-free
#pragma unroll
      for (int r = 0; r < TI / 2; ++r) {
        float t = aiv[r] + ajv;
        t = fmaxf(t, 0.f);
        acc[r] = fmaf(adjT[(ip * (TI / 2) + r) * JT + jj], t, acc[r]); // broadcast
      }
    }
  }
#pragma unroll
  for (int r = 0; r < TI / 2; ++r)
    S[((b * N_) + i0 + ip * (TI / 2) + r) * D_ + o] = acc[r];
}

// ---------------------------------------------------------------------------
// rowsum[m] = sum_j adj[m][j]   (one wave32 per row)
// ---------------------------------------------------------------------------
__global__ __launch_bounds__(256)
void rowsum_kernel(const float* __restrict__ adj, float* __restrict__ rowsum) {
  const int lane = threadIdx.x & 31;
  const int wave = threadIdx.x >> 5;
  const int row  = blockIdx.x * 8 + wave;
  const float* p = adj + row * N_;
  float s = 0.f;
  for (int j = lane; j < N_; j += 32) s += p[j];
#pragma unroll
  for (int off = 16; off > 0; off >>= 1) s += __shfl_down(s, off, 32);
  if (lane == 0) rowsum[row] = s;
}

// ---------------------------------------------------------------------------
extern "C" void kernel_launch(void* const* d_in, const int* in_sizes, int n_in,
                              void* d_out, int out_size, void* d_ws, size_t ws_size,
                              hipStream_t stream) {
  const float* node = (const float*)d_in[0];  // [B,N,D]
  const float* adj  = (const float*)d_in[1];  // [B,N,N]
  const float* w1   = (const float*)d_in[2];  // [L,D,2D]
  const float* b1   = (const float*)d_in[3];  // [L,D]
  const float* w2   = (const float*)d_in[4];  // [L,D,D]
  const float* b2   = (const float*)d_in[5];  // [L,D]

  float* h  = (float*)d_out;                  // running node features
  float* ai = (float*)d_ws;                   // [M, D]
  float* aj = ai + (size_t)M_ * D_;           // [M, D]
  float* S  = aj + (size_t)M_ * D_;           // [M, D]
  float* rs = S  + (size_t)M_ * D_;           // [M]

  hipMemcpyAsync(h, node, (size_t)M_ * D_ * sizeof(float),
                 hipMemcpyDeviceToDevice, stream);
  rowsum_kernel<<<M_ / 8, 256, 0, stream>>>(adj, rs);

  const dim3 gg(D_ / 16, M_ / 16);            // 8 x 128 C tiles
  for (int l = 0; l < L_; ++l) {
    const float* w1l = w1 + (size_t)l * D_ * 2 * D_;
    const float* b1l = b1 + (size_t)l * D_;
    const float* w2l = w2 + (size_t)l * D_ * D_;
    const float* b2l = b2 + (size_t)l * D_;
    // ai = h @ wi^T + b1   (wi = w1[l][:, :D]);  aj = h @ wj^T
    wmma_gemm_bias<<<gg, 32, 0, stream>>>(h, w1l, 2 * D_, 0,  b1l, ai);
    wmma_gemm_bias<<<gg, 32, 0, stream>>>(h, w1l, 2 * D_, D_, nullptr, aj);
    // S = sum_j adj * relu(ai + aj)
    pairwise_reduce<<<dim3(N_ / TI, B_), 256, 0, stream>>>(ai, aj, adj, S);
    // h += S @ w2^T + rowsum * b2
    wmma_gemm_final<<<gg, 32, 0, stream>>>(S, w2l, b2l, rs, h);
  }
}